// SynapticGNN_7748121002474
// MI455X (gfx1250) — compile-verified
//
#include <hip/hip_runtime.h>
#include <hip/hip_bf16.h>

#define NN 100000
#define NE 1600000
#define NSTRIPS (NN / 16)   // 6250, N divisible by 16

typedef __attribute__((ext_vector_type(2))) float v2f;
typedef __attribute__((ext_vector_type(8))) float v8f;

// ---------------- K1: deg init with self-loop weight 1.0 ----------------
__global__ void k_deg_init(float* __restrict__ deg) {
  int i = blockIdx.x * blockDim.x + threadIdx.x;
  if (i < NN) deg[i] = 1.0f;
}

// ---------------- K2: deg += ew at col ----------------
__global__ void k_deg_edges(const int* __restrict__ col,
                            const float* __restrict__ ew,
                            float* __restrict__ deg) {
  int e = blockIdx.x * blockDim.x + threadIdx.x;
  if (e < NE) atomicAdd(&deg[col[e]], ew[e]);
}

// ---------------- K3: dinv = deg>0 ? rsqrt(deg) : 0 ----------------
__global__ void k_dinv(const float* __restrict__ deg, float* __restrict__ dinv) {
  int i = blockIdx.x * blockDim.x + threadIdx.x;
  if (i < NN) {
    float d = deg[i];
    dinv[i] = (d > 0.0f) ? rsqrtf(d) : 0.0f;
  }
}

// ---------------- K4: H = X @ W  (N x 64) @ (64 x 64), fp32 WMMA ----------------
// One wave computes a 16-row x 64-col strip using V_WMMA_F32_16X16X4_F32,
// looping K in steps of 4 (16 steps), 4 column tiles per step.
__global__ void __launch_bounds__(256) k_gemm_x_w1(const float* __restrict__ X,
                                                   const float* __restrict__ W,
                                                   float* __restrict__ H) {
  const int lane  = threadIdx.x & 31;
  const int wave  = threadIdx.x >> 5;
  const int strip = blockIdx.x * 8 + wave;   // uniform per wave -> EXEC all-1 inside
  if (strip >= NSTRIPS) return;

  const int l    = lane & 15;
  const int half = lane >> 4;
  const int row0 = strip * 16;

  v8f acc0 = {}, acc1 = {}, acc2 = {}, acc3 = {};
  const float* __restrict__ xrow = X + (size_t)(row0 + l) * 64;

#pragma unroll
  for (int k0 = 0; k0 < 64; k0 += 4) {
    const int ka = k0 + 2 * half;          // lanes 0-15: K0/K1, lanes 16-31: K2/K3
    v2f a = *(const v2f*)(xrow + ka);      // 8B aligned (ka even)

    const float* __restrict__ w0 = W + (size_t)ka * 64;
    const float* __restrict__ w1 = W + (size_t)(ka + 1) * 64;
    v2f b0; b0.x = w0[ 0 + l]; b0.y = w1[ 0 + l];
    v2f b1; b1.x = w0[16 + l]; b1.y = w1[16 + l];
    v2f b2; b2.x = w0[32 + l]; b2.y = w1[32 + l];
    v2f b3; b3.x = w0[48 + l]; b3.y = w1[48 + l];

    acc0 = __builtin_amdgcn_wmma_f32_16x16x4_f32(false, a, false, b0, (short)0, acc0, false, false);
    acc1 = __builtin_amdgcn_wmma_f32_16x16x4_f32(false, a, false, b1, (short)0, acc1, false, false);
    acc2 = __builtin_amdgcn_wmma_f32_16x16x4_f32(false, a, false, b2, (short)0, acc2, false, false);
    acc3 = __builtin_amdgcn_wmma_f32_16x16x4_f32(false, a, false, b3, (short)0, acc3, false, false);
  }

  // C/D layout: VGPR v holds row v (lanes 0-15) / row v+8 (lanes 16-31)
#pragma unroll
  for (int v = 0; v < 8; ++v) {
    const size_t r = (size_t)(row0 + half * 8 + v) * 64;
    H[r +  0 + l] = acc0[v];
    H[r + 16 + l] = acc1[v];
    H[r + 32 + l] = acc2[v];
    H[r + 48 + l] = acc3[v];
  }
}

// ---------------- K5a: self-loop init  agg[i,:] = h[i,:]*dinv[i]^2 ----------------
__global__ void k_selfloop64(const float* __restrict__ h,
                             const float* __restrict__ dinv,
                             float* __restrict__ agg) {
  int idx = blockIdx.x * blockDim.x + threadIdx.x;     // NN*16 threads
  if (idx >= NN * 16) return;
  int node = idx >> 4;
  float s = dinv[node]; s = s * s;
  float4 v = ((const float4*)h)[idx];
  float4 o; o.x = v.x * s; o.y = v.y * s; o.z = v.z * s; o.w = v.w * s;
  ((float4*)agg)[idx] = o;
}

// ---------------- K5b: edge aggregate, 16 lanes per edge, float4 each ----------------
__global__ void k_edge_agg64(const int* __restrict__ rowi,
                             const int* __restrict__ coli,
                             const float* __restrict__ ew,
                             const float* __restrict__ dinv,
                             const float* __restrict__ h,
                             float* __restrict__ agg) {
  int idx = blockIdx.x * blockDim.x + threadIdx.x;     // NE*16 threads
  int e = idx >> 4;
  int q = idx & 15;
  if (e >= NE) return;
  int r = rowi[e];
  int c = coli[e];
  float norm = dinv[r] * ew[e] * dinv[c];
  float4 v = ((const float4*)h)[(size_t)r * 16 + q];
  float* dst = agg + (size_t)c * 64 + q * 4;
  atomicAdd(dst + 0, v.x * norm);
  atomicAdd(dst + 1, v.y * norm);
  atomicAdd(dst + 2, v.z * norm);
  atomicAdd(dst + 3, v.w * norm);
}

// ---------------- K6: h2[i] = dot(relu(agg[i,:]+b1), W2) ----------------
__global__ void k_relu_dot(const float* __restrict__ agg,
                           const float* __restrict__ b1,
                           const float* __restrict__ W2,
                           float* __restrict__ h2) {
  int i = blockIdx.x * blockDim.x + threadIdx.x;
  if (i >= NN) return;
  const float4* rowp = (const float4*)(agg + (size_t)i * 64);
  const float4* bp   = (const float4*)b1;
  const float4* wp   = (const float4*)W2;
  float s = 0.0f;
#pragma unroll
  for (int q = 0; q < 16; ++q) {
    float4 v = rowp[q];
    float4 b = bp[q];
    float4 w = wp[q];
    s += fmaxf(v.x + b.x, 0.0f) * w.x;
    s += fmaxf(v.y + b.y, 0.0f) * w.y;
    s += fmaxf(v.z + b.z, 0.0f) * w.z;
    s += fmaxf(v.w + b.w, 0.0f) * w.w;
  }
  h2[i] = s;
}

// ---------------- K7a: out[i] = h2[i]*dinv[i]^2 + b2 ----------------
__global__ void k_out_init(const float* __restrict__ h2,
                           const float* __restrict__ dinv,
                           const float* __restrict__ b2,
                           float* __restrict__ out) {
  int i = blockIdx.x * blockDim.x + threadIdx.x;
  if (i >= NN) return;
  float s = dinv[i];
  out[i] = h2[i] * s * s + b2[0];
}

// ---------------- K7b: out[c] += h2[r]*norm ----------------
__global__ void k_edge_out(const int* __restrict__ rowi,
                           const int* __restrict__ coli,
                           const float* __restrict__ ew,
                           const float* __restrict__ dinv,
                           const float* __restrict__ h2,
                           float* __restrict__ out) {
  int e = blockIdx.x * blockDim.x + threadIdx.x;
  if (e >= NE) return;
  int r = rowi[e];
  int c = coli[e];
  float norm = dinv[r] * ew[e] * dinv[c];
  atomicAdd(&out[c], h2[r] * norm);
}

extern "C" void kernel_launch(void* const* d_in, const int* in_sizes, int n_in,
                              void* d_out, int out_size, void* d_ws, size_t ws_size,
                              hipStream_t stream) {
  const float* x    = (const float*)d_in[0];              // [NN,64]
  const int*   ei   = (const int*)d_in[1];                // [2,NE] flat
  const float* ew   = (const float*)d_in[2];              // [NE]
  const float* W1   = (const float*)d_in[3];              // [64,64]
  const float* b1   = (const float*)d_in[4];              // [64]
  const float* W2   = (const float*)d_in[5];              // [64,1]
  const float* b2   = (const float*)d_in[6];              // [1]
  float* out = (float*)d_out;                             // [NN]

  const int* rowi = ei;        // edge_index[0]
  const int* coli = ei + NE;   // edge_index[1]

  // workspace carve-up (all offsets 16B aligned)
  char* ws = (char*)d_ws;
  float* h1   = (float*)(ws);                                   // NN*64
  float* agg1 = (float*)(ws + (size_t)NN * 64 * 4);             // NN*64
  float* deg  = (float*)(ws + (size_t)NN * 128 * 4);            // NN
  float* dinv = (float*)(ws + (size_t)NN * 129 * 4);            // NN
  float* h2   = (float*)(ws + (size_t)NN * 130 * 4);            // NN

  const int T = 256;
  // degree + norm
  k_deg_init <<<(NN + T - 1) / T, T, 0, stream>>>(deg);
  k_deg_edges<<<(NE + T - 1) / T, T, 0, stream>>>(coli, ew, deg);
  k_dinv     <<<(NN + T - 1) / T, T, 0, stream>>>(deg, dinv);

  // layer 1: GEMM (WMMA fp32) then aggregate
  k_gemm_x_w1<<<(NSTRIPS + 7) / 8, T, 0, stream>>>(x, W1, h1);
  k_selfloop64<<<(NN * 16 + T - 1) / T, T, 0, stream>>>(h1, dinv, agg1);
  k_edge_agg64<<<(NE * 16 + T - 1) / T, T, 0, stream>>>(rowi, coli, ew, dinv, h1, agg1);

  // layer 2: bias+ReLU fused into per-node dot with W2, then scalar aggregate
  k_relu_dot<<<(NN + T - 1) / T, T, 0, stream>>>(agg1, b1, W2, h2);
  k_out_init<<<(NN + T - 1) / T, T, 0, stream>>>(h2, dinv, b2, out);
  k_edge_out<<<(NE + T - 1) / T, T, 0, stream>>>(rowi, coli, ew, dinv, h2, out);
}